// AE_43851616092249
// MI455X (gfx1250) — compile-verified
//
#include <hip/hip_runtime.h>
#include <hip/hip_bf16.h>

typedef __attribute__((ext_vector_type(2))) float v2f;
typedef __attribute__((ext_vector_type(8))) float v8f;

// swap values between lane l and lane l^16 (wave32)
static __device__ __forceinline__ float lane_xor16(float v) {
    return __shfl_xor(v, 16, 32);
}

// Branch-free tanh using the hardware transcendentals:
// tanh(x) = sign(x) * (1 - e) / (1 + e),  e = exp(-2|x|) = exp2(-2*log2(e)*|x|)
// |x| large -> e == 0 -> result +-1; no EXEC divergence, ~2-3 ulp.
static __device__ __forceinline__ float fast_tanh(float v) {
    const float a = __builtin_fabsf(v);
    const float e = __builtin_amdgcn_exp2f(a * -2.8853900817779268f);
    const float r = (1.0f - e) * __builtin_amdgcn_rcpf(1.0f + e);
    return __builtin_copysignf(r, v);
}

__global__ __launch_bounds__(256) void ae_rosenbrock_kernel(
    const float* __restrict__ x,
    const float* __restrict__ W1,   // [2,16] row-major
    const float* __restrict__ b1,   // [16]
    const float* __restrict__ W2,   // [16,2] row-major
    const float* __restrict__ b2,   // [2]
    float* __restrict__ out,        // [rows,2]
    int rows)
{
    const int  lane   = threadIdx.x & 31;
    const int  wave   = (int)((blockIdx.x * blockDim.x + threadIdx.x) >> 5);
    const int  nWaves = (int)((gridDim.x * blockDim.x) >> 5);
    const int  m      = lane & 15;
    const bool hi     = (lane >= 16);
    const int  nChunks = rows >> 5;            // 32 rows per wave-chunk

    // ---------- chunk-invariant WMMA operands ----------
    // A1 = W1^T as 16x4 f32 A-operand (real K=2, K=2..3 zero padded).
    const float w1r0 = W1[m];        // W1[0][m]
    const float w1r1 = W1[16 + m];   // W1[1][m]
    v2f a1;
    a1.x = hi ? 0.0f : w1r0;
    a1.y = hi ? 0.0f : w1r1;

    // Layer-1 bias folded into the WMMA C operand:
    // C layout: VGPR v holds M = v + 8*hi for every lane -> C[v] = b1[v + 8*hi].
    v8f cbias1;
    {
        const int hb = hi ? 8 : 0;
        #pragma unroll
        for (int v = 0; v < 8; ++v) cbias1[v] = b1[hb + v];
    }

    // A2 chunks: W2^T (2x16) split into four 16x4 K-chunks. Only M<2 lanes carry data.
    v2f a2[4];
    {
        const int  mm    = (m < 2) ? m : 1;   // clamp so loads stay in-bounds
        const bool valid = (m < 2);
        #pragma unroll
        for (int c = 0; c < 4; ++c) {
            const int kk0 = 4 * c + (hi ? 2 : 0);
            const float wa = W2[kk0 * 2 + mm];        // W2^T[m][kk0]
            const float wb = W2[(kk0 + 1) * 2 + mm];  // W2^T[m][kk0+1]
            a2[c].x = valid ? wa : 0.0f;
            a2[c].y = valid ? wb : 0.0f;
        }
    }

    // Layer-2 bias folded into the first accumulator: D2[m][r] += b2[m], m=0,1.
    v8f cbias2 = {};
    cbias2[0] = b2[0];
    cbias2[1] = b2[1];

    const float2* __restrict__ x2 = (const float2*)x;
    float2* __restrict__ o2 = (float2*)out;

    int chunk = wave;
    if (chunk >= nChunks) return;              // wave-uniform

    float2 xv = x2[(chunk << 5) + lane];       // prologue load

    while (true) {
        const int row = (chunk << 5) + lane;   // this lane's batch row

        // B1 operands for the two 16-row tiles (lane=N=row; lanes 16-31 = K2,K3 pad).
        const float xs0 = lane_xor16(xv.x);
        const float xs1 = lane_xor16(xv.y);
        v2f bT[2];
        bT[0].x = hi ? 0.0f : xv.x;  bT[0].y = hi ? 0.0f : xv.y;  // rows +0..15
        bT[1].x = hi ? 0.0f : xs0;   bT[1].y = hi ? 0.0f : xs1;   // rows +16..31

        float z0t[2], z1t[2];
        #pragma unroll
        for (int tile = 0; tile < 2; ++tile) {
            // ---- layer 1: H^T = W1^T @ X^T + b1  (16x16 f32, bias via C) ----
            v8f d = __builtin_amdgcn_wmma_f32_16x16x4_f32(
                        false, a1, false, bT[tile], (short)0, cbias1, false, false);

            // lane l<16 holds H^T[0..7][row l]; lane l+16 holds H^T[8..15][row l].
            float t[8], xs[8];
            #pragma unroll
            for (int v = 0; v < 8; ++v) t[v] = fast_tanh(d[v]);
            #pragma unroll
            for (int v = 0; v < 8; ++v) xs[v] = lane_xor16(t[v]);

            // ---- layer 2: Z^T = W2^T @ H^T + b2, K=16 as four chained K=4 WMMAs ----
            v2f bh[4];
            bh[0].x = hi ? xs[2] : t[0];  bh[0].y = hi ? xs[3] : t[1];
            bh[1].x = hi ? xs[6] : t[4];  bh[1].y = hi ? xs[7] : t[5];
            bh[2].x = hi ? t[2] : xs[0];  bh[2].y = hi ? t[3] : xs[1];
            bh[3].x = hi ? t[6] : xs[4];  bh[3].y = hi ? t[7] : xs[5];

            v8f acc = cbias2;
            #pragma unroll
            for (int c = 0; c < 4; ++c)
                acc = __builtin_amdgcn_wmma_f32_16x16x4_f32(
                          false, a2[c], false, bh[c], (short)0, acc, false, false);
            z0t[tile] = acc[0];   // z[row][0], valid at lanes 0-15
            z1t[tile] = acc[1];   // z[row][1], valid at lanes 0-15
        }

        // redistribute: lanes 0-15 keep tile0 rows, lanes 16-31 take tile1 rows.
        const float z0s = lane_xor16(z0t[1]);
        const float z1s = lane_xor16(z1t[1]);
        const float z0 = hi ? z0s : z0t[0];
        const float z1 = hi ? z1s : z1t[0];

        // software pipeline: kick off next chunk's x load; the decoder loop hides it.
        const int  next     = chunk + nWaves;
        const bool has_next = (next < nChunks);           // wave-uniform
        float2 xnext;
        if (has_next) xnext = x2[(next << 5) + lane];

        // ---- decoder: 100 steps of quadratic-penalty gradient descent ----
        float p0 = z0, p1 = z1;
        #pragma unroll 4
        for (int it = 0; it < 100; ++it) {
            const float e0 = p0 - 1.0f;
            const float c1 = p1 - 1.0f - e0 * e0;   // >= 0 feasible
            const float c2 = 2.0f - p0 - p1;        // >= 0 feasible
            const float v1 = fminf(c1, 0.0f);
            const float v2 = fminf(c2, 0.0f);
            const float g0 = (p0 - z0) + 20.0f * (v1 * (-2.0f * e0) - v2); // 2*RHO
            const float g1 = (p1 - z1) + 20.0f * (v1 - v2);
            p0 -= 0.05f * g0;                        // LR = 0.05
            p1 -= 0.05f * g1;
        }

        float2 r; r.x = p0; r.y = p1;
        o2[row] = r;

        if (!has_next) break;
        chunk = next;
        xv = xnext;
    }
}

extern "C" void kernel_launch(void* const* d_in, const int* in_sizes, int n_in,
                              void* d_out, int out_size, void* d_ws, size_t ws_size,
                              hipStream_t stream) {
    const float* x  = (const float*)d_in[0];
    const float* W1 = (const float*)d_in[1];
    const float* b1 = (const float*)d_in[2];
    const float* W2 = (const float*)d_in[3];
    const float* b2 = (const float*)d_in[4];
    float* out = (float*)d_out;

    const int rows = in_sizes[0] / 2;   // [B,2] flat
    const int blocks = 2048;            // 16384 waves, grid-stride over 32768 chunks
    ae_rosenbrock_kernel<<<blocks, 256, 0, stream>>>(x, W1, b1, W2, b2, out, rows);
}